// GMPolicy_2611340116240
// MI455X (gfx1250) — compile-verified
//
#include <hip/hip_runtime.h>
#include <cstdint>
#include <cstddef>

#define GM_EPS 1e-4f

namespace {

constexpr int kB          = 2;
constexpr int kK          = 8;
constexpr int kC          = 16;
constexpr int kS          = 12 * 60 * 104;   // 74880 spatial points (F*H*W)
constexpr int kTile       = 128;             // s-points handled per workgroup
constexpr int kRows       = kK * kC;         // 128 mu rows per batch
constexpr int kRowsPerWave = kRows / 4;      // 32 rows per wave's TDM descriptor
constexpr int kBlocksPerB = kS / kTile;      // 585 (divides exactly)

typedef unsigned u32;
typedef __attribute__((ext_vector_type(4))) u32 su32x4;   // D# group 0 (4 SGPRs)
typedef __attribute__((ext_vector_type(8))) u32 su32x8;   // D# group 1 (8 SGPRs)

// Byte offset of a __shared__ object within the workgroup LDS allocation.
__device__ __forceinline__ unsigned lds_addr_of(const void* p) {
  return (unsigned)(size_t)((__attribute__((address_space(3))) const void*)p);
}

__global__ __launch_bounds__(kTile)
void gm_policy_fused(const float* __restrict__ x_t,
                     const float* __restrict__ sigma_t,
                     const float* __restrict__ a_s,
                     const float* __restrict__ mu_s,
                     const float* __restrict__ std_s,
                     const float* __restrict__ x_s,
                     const float* __restrict__ sigma_s,
                     float* __restrict__ out) {
  __shared__ float lds_mu[kRows * kTile];   // 128 rows x 512 B = 64 KB

  const int tid = threadIdx.x;
  const int b   = blockIdx.x / kBlocksPerB;
  const int s0  = (blockIdx.x % kBlocksPerB) * kTile;
  const int s   = s0 + tid;

  // ---------------------------------------------------------------
  // Phase 0: each of the 4 waves issues ONE Tensor Data Mover
  // descriptor for its 32-row slice of the mu tile (32 x 128 floats,
  // row stride kS floats, 16 KB into LDS). TDM ignores EXEC and
  // issues per wave, which is exactly what we want here: 4 DMAs in
  // flight, each wave later waits on its own TENSORcnt. All D#
  // inputs are wave-uniform, so the descriptor lives in SGPRs.
  // ---------------------------------------------------------------
  {
    const u32 w = __builtin_amdgcn_readfirstlane((unsigned)tid) >> 5;  // wave id 0..3
    const uint64_t gaddr = (uint64_t)((const char*)mu_s +
        (((size_t)(b * kRows + (int)w * kRowsPerWave) * kS + (size_t)s0) << 2));
    const u32 lds_base = lds_addr_of(lds_mu) + w * (kRowsPerWave * kTile * 4u);

    // D# group 0: control / lds_addr / 57-bit global tile address / type=2
    const su32x4 g0 = {
        0x1u,                                           // count=1, user D#
        lds_base,                                       // lds_addr (bytes)
        (u32)gaddr,                                     // global_addr[31:0]
        ((u32)(gaddr >> 32) & 0x01FFFFFFu) | (2u << 30) // global_addr[56:32] | type=2
    };
    // D# group 1: sizes/strides (data_size=4B code 2; dims in elements)
    const su32x8 g1 = {
        0x00020000u,                                  // wg_mask=0 | data_size=2 (4 B)
        ((u32)kS & 0xFFFFu) << 16,                    // tensor_dim0[15:0] (kS=0x12480)
        ((u32)kS >> 16) | ((u32)kRowsPerWave << 16),  // tensor_dim0[31:16] | tensor_dim1=32
        (128u << 16),                                 // tensor_dim1 hi=0 | tile_dim0=128
        (u32)kRowsPerWave,                            // tile_dim1=32 | tile_dim2=0
        (u32)kS,                                      // tensor_dim0_stride[31:0]
        0u,                                           // stride0[47:32]=0 | dim1_stride lo=0
        0u                                            // tensor_dim1_stride hi=0
    };
    asm volatile("tensor_load_to_lds %0, %1" :: "s"(g0), "s"(g1) : "memory");
  }

  // ------- overlap with the DMA: scalars + per-c / per-k loads + prep -------
  const float st  = sigma_t[b];
  const float ss  = sigma_s[b];
  const float sdv = std_s[b];
  const bool all_eq = (sigma_s[0] == sigma_t[0]) && (sigma_s[1] == sigma_t[1]);

  const float st2 = st * st;
  const float ss2 = ss * ss;
  const float omt = 1.0f - st;
  const float oms = 1.0f - ss;
  const float xi_raw = ss2 * omt * omt - st2 * oms * oms;
  const float xi_x   = fmaxf(xi_raw, GM_EPS);
  const float inv_xi = 1.0f / xi_x;
  const float x0var_t = ss2 * st2 * inv_xi;
  const float x0var_s = (sdv * ss) * (sdv * ss);
  const float var_sum = x0var_t + x0var_s;
  const float denomi  = fmaxf(var_sum, GM_EPS);
  const float inv_den = 1.0f / denomi;
  const bool  is_safe = (xi_raw > GM_EPS) || (var_sum > GM_EPS);
  const float inv_st  = 1.0f / fmaxf(st, GM_EPS);
  const bool  use_vs  = all_eq || !is_safe;
  // x0_t[c] = (ss^2 (1-st) x_t - st^2 (1-ss) x_s) / xi
  const float cA =  ss2 * omt * inv_xi;
  const float cB = -(st2 * oms) * inv_xi;

  float xt[kC], xs[kC];
  {
    const float* pt = x_t + (size_t)b * kC * kS + s;
    const float* ps = x_s + (size_t)b * kC * kS + s;
#pragma unroll
    for (int c = 0; c < kC; ++c) {
      xt[c] = pt[(size_t)c * kS];
      xs[c] = ps[(size_t)c * kS];
    }
  }

  float a[kK];
  {
    const float* pa = a_s + (size_t)b * kK * kS + s;
#pragma unroll
    for (int k = 0; k < kK; ++k) a[k] = pa[(size_t)k * kS];
  }

  // softmax over K of a_s
  float As[kK];
  {
    float m = a[0];
#pragma unroll
    for (int k = 1; k < kK; ++k) m = fmaxf(m, a[k]);
    float sum = 0.0f;
#pragma unroll
    for (int k = 0; k < kK; ++k) { As[k] = __expf(a[k] - m); sum += As[k]; }
    const float inv = 1.0f / sum;
#pragma unroll
    for (int k = 0; k < kK; ++k) As[k] *= inv;
  }

  // x0_t, d = x_s - x0_t, and Σ d^2 (for the expanded distance)
  float x0t[kC], dd[kC];
  float sumd2 = 0.0f;
#pragma unroll
  for (int c = 0; c < kC; ++c) {
    x0t[c] = cA * xt[c] + cB * xs[c];
    dd[c]  = xs[c] - x0t[c];
    sumd2  = fmaf(dd[c], dd[c], sumd2);
  }

  // ---- DMA completion (own slice) + workgroup visibility (all slices) ----
#if __has_builtin(__builtin_amdgcn_s_wait_tensorcnt)
  __builtin_amdgcn_s_wait_tensorcnt(0);
#else
  asm volatile("s_wait_tensorcnt 0x0" ::: "memory");
#endif
  __syncthreads();

  // ---------------- Pass 1 over LDS tile: v_s, p[k], q[k] ----------------
  // dist[k] = Σc (d[c] - ss*mu)^2 = sumd2 - 2 ss p[k] + ss^2 q[k]
  float vs[kC];
#pragma unroll
  for (int c = 0; c < kC; ++c) vs[c] = 0.0f;

  float dist[kK];
#pragma unroll
  for (int k = 0; k < kK; ++k) {
    float p = 0.0f, q = 0.0f;
#pragma unroll
    for (int c = 0; c < kC; ++c) {
      const float m = lds_mu[(k * kC + c) * kTile + tid];
      vs[c] = fmaf(As[k], m, vs[c]);
      p = fmaf(dd[c], m, p);
      q = fmaf(m, m, q);
    }
    dist[k] = sumd2 - 2.0f * ss * p + ss2 * q;
  }

  // softmax over K of a_t = a_s - 0.5*dist/denomi
  float At[kK];
  float inv_tsum;
  {
    float m = -3.4e38f;
#pragma unroll
    for (int k = 0; k < kK; ++k) {
      At[k] = a[k] - 0.5f * dist[k] * inv_den;
      m = fmaxf(m, At[k]);
    }
    float sum = 0.0f;
#pragma unroll
    for (int k = 0; k < kK; ++k) { At[k] = __expf(At[k] - m); sum += At[k]; }
    inv_tsum = 1.0f / sum;
  }

  // ---------------- Pass 2 over LDS tile: x_0, v_t, store ----------------
  float* po = out + (size_t)b * kC * kS + s;
#pragma unroll
  for (int c = 0; c < kC; ++c) {
    float mt = 0.0f;
#pragma unroll
    for (int k = 0; k < kK; ++k)
      mt = fmaf(At[k], lds_mu[(k * kC + c) * kTile + tid], mt);
    mt *= inv_tsum;                 // Σk A_t[k] * mu[k,c]
    // x_0 = (x0var_t * (x_s - ss*mt) + x0var_s * x0_t) / denomi   (ΣA_t = 1)
    const float x0c = inv_den * (x0var_t * (xs[c] - ss * mt) + x0var_s * x0t[c]);
    const float vt  = (xt[c] - x0c) * inv_st;
    po[(size_t)c * kS] = use_vs ? vs[c] : vt;
  }
}

}  // namespace

extern "C" void kernel_launch(void* const* d_in, const int* in_sizes, int n_in,
                              void* d_out, int out_size, void* d_ws, size_t ws_size,
                              hipStream_t stream) {
  const float* x_t     = (const float*)d_in[0];
  const float* sigma_t = (const float*)d_in[1];
  const float* a_s     = (const float*)d_in[2];
  const float* mu_s    = (const float*)d_in[3];
  const float* std_s   = (const float*)d_in[4];
  const float* x_s     = (const float*)d_in[5];
  const float* sigma_s = (const float*)d_in[6];
  float* out           = (float*)d_out;

  dim3 grid(kB * kBlocksPerB);   // 1170 blocks, exact cover of B*S
  dim3 block(kTile);             // 128 threads = 4 waves (wave32)
  gm_policy_fused<<<grid, block, 0, stream>>>(x_t, sigma_t, a_s, mu_s,
                                              std_s, x_s, sigma_s, out);
}